// NestedLogitManager_13134009991482
// MI455X (gfx1250) — compile-verified
//
#include <hip/hip_runtime.h>
#include <math.h>

// ---- problem constants -----------------------------------------------------
#define BB 512
#define TT 2048
#define MM 16
#define SS 64
#define HH 64
#define EE 32
#define NEGV (-1e10f)

typedef float v2f __attribute__((ext_vector_type(2)));
typedef float v8f __attribute__((ext_vector_type(8)));

// ---- fp32 WMMA 16x16x4 helpers --------------------------------------------
// A: 16xK row-major in LDS (stride lda, even).
// Wsw: KxN weights pre-swizzled in LDS so the (k,k+1) row pair of a column is
//      contiguous: Wsw[((row>>1)*N + col)*2 + (row&1)]. Since k+h is even, the
//      B fragment is ONE aligned 8-byte LDS load (no cross-row packing movs).
// Fragment mapping (CDNA5 ISA 7.12.2):
//   A lane l: M = l&15, K = 4*kstep + 2*(l>>4) + vgpr
//   B lane l: N = l&15, K = 4*kstep + 2*(l>>4) + vgpr
//   C lane l, vgpr v: M = v + 8*(l>>4), N = l&15
__device__ __forceinline__ v8f wmma_tile_f32(const float* A, int lda,
                                             const float* Wsw, int N,
                                             int K, int n0, int li, int h) {
  v8f c = {0.f, 0.f, 0.f, 0.f, 0.f, 0.f, 0.f, 0.f};
#pragma unroll
  for (int k = 0; k < K; k += 4) {
    v2f a = *(const v2f*)&A[li * lda + k + h];                 // contiguous pair
    v2f b = *(const v2f*)&Wsw[(k + h) * N + 2 * (n0 + li)];    // swizzled pair
    c = __builtin_amdgcn_wmma_f32_16x16x4_f32(false, a, false, b, (short)0, c,
                                              false, false);
  }
  return c;
}

// stage KxN weights from global into LDS in the swizzled pair layout
__device__ __forceinline__ void stage_w_swizzled(float* dst,
                                                 const float* __restrict__ src,
                                                 int K, int N, int tid,
                                                 int nthr) {
  for (int i = tid; i < K * N; i += nthr) {
    int row = i / N, col = i - row * N;
    dst[((row >> 1) * N + col) * 2 + (row & 1)] = src[i];
  }
}

__device__ __forceinline__ void store_c_tile(v8f c, float* Y, int ldy, int n0,
                                             const float* bias, int li, int h,
                                             bool do_relu) {
  int r0 = (h >> 1) * 8;  // h in {0,2} -> row base 0 or 8
  float bv = bias[n0 + li];
#pragma unroll
  for (int v = 0; v < 8; ++v) {
    float val = c[v] + bv;
    if (do_relu) val = fmaxf(val, 0.f);
    Y[(r0 + v) * ldy + n0 + li] = val;
  }
}

// ---- K0: eta ---------------------------------------------------------------
__global__ void k_eta(const float* __restrict__ lam, float* __restrict__ eta) {
  int m = threadIdx.x;
  if (m < MM) {
    float e = 1.f / (1.f + __expf(-lam[m]));
    eta[m] = fminf(fmaxf(e, 0.1f), 1.f);
  }
}

// ---- K1: utilities MLP ([state|embed] 96 -> 64 -> 64 -> 1) -----------------
__global__ __launch_bounds__(64) void k_utilities(
    const float* __restrict__ state, const float* __restrict__ tf,
    const float* __restrict__ Wte, const float* __restrict__ bte,
    const float* __restrict__ W1, const float* __restrict__ b1,
    const float* __restrict__ W2, const float* __restrict__ b2,
    const float* __restrict__ W3, const float* __restrict__ b3,
    float* __restrict__ out_util) {
  __shared__ __attribute__((aligned(16))) float sW1[96 * 64];
  __shared__ __attribute__((aligned(16))) float sW2[64 * 64];
  __shared__ float sW3[64], sB1[64], sB2[64], sWte[5 * 32], sBte[32];
  __shared__ __attribute__((aligned(16))) float sX[2][16][96];
  __shared__ __attribute__((aligned(16))) float sY[2][16][64];

  const int tid = threadIdx.x;
  stage_w_swizzled(sW1, W1, 96, 64, tid, 64);
  stage_w_swizzled(sW2, W2, 64, 64, tid, 64);
  if (tid < 64) { sW3[tid] = W3[tid]; sB1[tid] = b1[tid]; sB2[tid] = b2[tid]; }
  for (int i = tid; i < 160; i += 64) sWte[i] = Wte[i];
  if (tid < 32) sBte[tid] = bte[tid];
  __syncthreads();

  const int lane = tid & 31, wave = tid >> 5;
  const int li = lane & 15, h = (lane >> 4) << 1;  // 0 or 2
  const float b3v = b3[0];

  for (int it = 0; it < 8; ++it) {
    int tile = (blockIdx.x * 2 + wave) + it * 8192;  // 65536 tiles total
    int row0 = tile << 4;
    int g = row0 + li;
    int bi = g >> 11;  // /T
    if (h == 0) {  // lanes 0..15: copy state row (64 f32 = 16 float4)
      const float4* sp = (const float4*)(state + bi * SS);
      float4* dp = (float4*)&sX[wave][li][0];
#pragma unroll
      for (int j = 0; j < 16; ++j) dp[j] = sp[j];
    } else {  // lanes 16..31: compute task embedding (5 -> 32)
      float t0 = tf[g * 5 + 0], t1 = tf[g * 5 + 1], t2 = tf[g * 5 + 2];
      float t3 = tf[g * 5 + 3], t4 = tf[g * 5 + 4];
#pragma unroll
      for (int e = 0; e < EE; ++e) {
        sX[wave][li][64 + e] = sBte[e] + t0 * sWte[e] + t1 * sWte[32 + e] +
                               t2 * sWte[64 + e] + t3 * sWte[96 + e] +
                               t4 * sWte[128 + e];
      }
    }
    // layer1: 96 -> 64, relu
#pragma unroll
    for (int n = 0; n < 4; ++n) {
      v8f c = wmma_tile_f32(&sX[wave][0][0], 96, sW1, 64, 96, n * 16, li, h);
      store_c_tile(c, &sY[wave][0][0], 64, n * 16, sB1, li, h, true);
    }
    // layer2: 64 -> 64, relu (write into sX cols 0..63)
#pragma unroll
    for (int n = 0; n < 4; ++n) {
      v8f c = wmma_tile_f32(&sY[wave][0][0], 64, sW2, 64, 64, n * 16, li, h);
      store_c_tile(c, &sX[wave][0][0], 96, n * 16, sB2, li, h, true);
    }
    // layer3: 64 -> 1 dot product, lanes 0..15 own one row each
    if (h == 0) {
      float acc = b3v;
#pragma unroll
      for (int j = 0; j < 64; ++j) acc += sX[wave][li][j] * sW3[j];
      out_util[row0 + li] = acc;
    }
  }
}

// ---- K2a: gctx = relu(state@Wg1+bg1)@Wg2+bg2  (512 rows, 64->64->32) ------
__global__ __launch_bounds__(64) void k_gctx(
    const float* __restrict__ state, const float* __restrict__ Wg1,
    const float* __restrict__ bg1, const float* __restrict__ Wg2,
    const float* __restrict__ bg2, float* __restrict__ gctx) {
  __shared__ __attribute__((aligned(16))) float sW1[64 * 64];
  __shared__ __attribute__((aligned(16))) float sW2[64 * 32];
  __shared__ float sB1[64], sB2[32];
  __shared__ __attribute__((aligned(16))) float sX[2][16][64];
  __shared__ __attribute__((aligned(16))) float sY[2][16][64];
  const int tid = threadIdx.x;
  stage_w_swizzled(sW1, Wg1, 64, 64, tid, 64);
  stage_w_swizzled(sW2, Wg2, 64, 32, tid, 64);
  if (tid < 64) sB1[tid] = bg1[tid];
  if (tid < 32) sB2[tid] = bg2[tid];
  __syncthreads();

  const int lane = tid & 31, wave = tid >> 5;
  const int li = lane & 15, h = (lane >> 4) << 1;
  int tile = blockIdx.x * 2 + wave;  // 32 tiles
  int row0 = tile << 4;
  // cooperative copy of 16x64 state tile
  {
    const float4* sp = (const float4*)(state + row0 * SS);
    float4* dp = (float4*)&sX[wave][0][0];
    for (int i = lane; i < 16 * 64 / 4; i += 32) dp[i] = sp[i];
  }
#pragma unroll
  for (int n = 0; n < 4; ++n) {
    v8f c = wmma_tile_f32(&sX[wave][0][0], 64, sW1, 64, 64, n * 16, li, h);
    store_c_tile(c, &sY[wave][0][0], 64, n * 16, sB1, li, h, true);
  }
  int r0 = (h >> 1) * 8;
#pragma unroll
  for (int n = 0; n < 2; ++n) {
    v8f c = wmma_tile_f32(&sY[wave][0][0], 64, sW2, 32, 64, n * 16, li, h);
    float bv = sB2[n * 16 + li];
#pragma unroll
    for (int v = 0; v < 8; ++v)
      gctx[(row0 + r0 + v) * EE + n * 16 + li] = c[v] + bv;
  }
}

// ---- K2b: masked mean/max pool of (recomputed) embeddings + W_a MLP -------
// one block per (b, m); embeds recomputed from L2-resident task_features
__global__ __launch_bounds__(256) void k_pool(
    const float* __restrict__ tf, const float* __restrict__ Wte,
    const float* __restrict__ bte, const int* __restrict__ mask,
    const float* __restrict__ Wa, const float* __restrict__ ba,
    float* __restrict__ pooled) {
  int b = blockIdx.x >> 4, m = blockIdx.x & 15;
  int tid = threadIdx.x, e = tid & 31, wv = tid >> 5;
  float w0 = Wte[0 * EE + e], w1 = Wte[1 * EE + e], w2 = Wte[2 * EE + e];
  float w3 = Wte[3 * EE + e], w4 = Wte[4 * EE + e], be = bte[e];
  const int* mrow = mask + (b * MM + m) * TT;
  const float* tfb = tf + (size_t)b * TT * 5;
  float sum = 0.f, mx = -INFINITY, cnt = 0.f;
  bool anyz = false;
  for (int t = wv; t < TT; t += 8) {
    int msk = mrow[t];
    if (msk) {
      float t0 = tfb[t * 5 + 0], t1 = tfb[t * 5 + 1], t2 = tfb[t * 5 + 2];
      float t3 = tfb[t * 5 + 3], t4 = tfb[t * 5 + 4];
      float emb = be + t0 * w0 + t1 * w1 + t2 * w2 + t3 * w3 + t4 * w4;
      sum += emb;
      mx = fmaxf(mx, emb);
      cnt += 1.f;
    } else {
      anyz = true;  // masked-out entries contribute 0 to the max (ref: me.max)
    }
  }
  if (anyz) mx = fmaxf(mx, 0.f);

  __shared__ float rs[8][32], rm[8][32], rc[8];
  __shared__ float cat[64];
  rs[wv][e] = sum;
  rm[wv][e] = mx;
  if (e == 0) rc[wv] = cnt;
  __syncthreads();
  if (tid < 32) {
    float s = 0.f, g = -INFINITY, c = 0.f;
#pragma unroll
    for (int w = 0; w < 8; ++w) {
      s += rs[w][e];
      g = fmaxf(g, rm[w][e]);
      c += rc[w];
    }
    cat[e] = s / fmaxf(c, 1.f);  // mean pool
    cat[32 + e] = g;             // max pool
  }
  __syncthreads();
  if (tid < 32) {
    float acc = ba[e];
#pragma unroll
    for (int j = 0; j < 64; ++j) acc += cat[j] * Wa[j * EE + e];
    pooled[(b * MM + m) * EE + e] = fmaxf(acc, 0.f);
  }
}

// ---- K2c: nest-score MLP ([gctx|pooled|eye] 80 -> 64 -> 64 -> 1) ----------
__global__ __launch_bounds__(64) void k_nest_scores(
    const float* __restrict__ gctx, const float* __restrict__ pooled,
    const float* __restrict__ W1, const float* __restrict__ b1,
    const float* __restrict__ W2, const float* __restrict__ b2,
    const float* __restrict__ W3, const float* __restrict__ b3,
    float* __restrict__ out_ns) {
  __shared__ __attribute__((aligned(16))) float sW1[80 * 64];
  __shared__ __attribute__((aligned(16))) float sW2[64 * 64];
  __shared__ float sW3[64], sB1[64], sB2[64];
  __shared__ __attribute__((aligned(16))) float sX[2][16][80];
  __shared__ __attribute__((aligned(16))) float sY[2][16][64];
  const int tid = threadIdx.x;
  stage_w_swizzled(sW1, W1, 80, 64, tid, 64);
  stage_w_swizzled(sW2, W2, 64, 64, tid, 64);
  if (tid < 64) { sW3[tid] = W3[tid]; sB1[tid] = b1[tid]; sB2[tid] = b2[tid]; }
  __syncthreads();

  const int lane = tid & 31, wave = tid >> 5;
  const int li = lane & 15, h = (lane >> 4) << 1;
  int b = blockIdx.x * 2 + wave;  // tile == batch index (rows are m=0..15)
  // build input rows: [gctx(32) | pooled(32) | onehot(16)]
  for (int idx = lane; idx < 16 * 80; idx += 32) {
    int r = idx / 80, c = idx % 80;
    float v;
    if (c < 32) v = gctx[b * EE + c];
    else if (c < 64) v = pooled[(b * MM + r) * EE + (c - 32)];
    else v = ((c - 64) == r) ? 1.f : 0.f;
    sX[wave][r][c] = v;
  }
#pragma unroll
  for (int n = 0; n < 4; ++n) {
    v8f c = wmma_tile_f32(&sX[wave][0][0], 80, sW1, 64, 80, n * 16, li, h);
    store_c_tile(c, &sY[wave][0][0], 64, n * 16, sB1, li, h, true);
  }
#pragma unroll
  for (int n = 0; n < 4; ++n) {
    v8f c = wmma_tile_f32(&sY[wave][0][0], 64, sW2, 64, 64, n * 16, li, h);
    store_c_tile(c, &sX[wave][0][0], 80, n * 16, sB2, li, h, true);
  }
  if (h == 0) {
    float acc = b3[0];
#pragma unroll
    for (int j = 0; j < 64; ++j) acc += sX[wave][li][j] * sW3[j];
    out_ns[b * MM + li] = acc;
  }
}

// ---- K3: per-(b,m) masked max + logsumexp; save max & Z for softmax reuse -
__global__ __launch_bounds__(256) void k_inclusive(
    const float* __restrict__ util, const int* __restrict__ mask,
    const float* __restrict__ eta, float* __restrict__ maxu_ws,
    float* __restrict__ Z_ws, float* __restrict__ incl_out) {
  int bm = blockIdx.x;               // == b*16 + m
  int b = bm >> 4, m = bm & 15;
  int tid = threadIdx.x;
  float ieta = 1.f / eta[m];
  const int* mrow = mask + (size_t)bm * TT;
  const float* urow = util + (size_t)b * TT;
  float mu[8];
  float mx = -INFINITY;
#pragma unroll
  for (int i = 0; i < 8; ++i) {
    int t = tid + i * 256;
    mu[i] = mrow[t] ? urow[t] : NEGV;
    mx = fmaxf(mx, mu[i]);
  }
#pragma unroll
  for (int off = 16; off > 0; off >>= 1) mx = fmaxf(mx, __shfl_xor(mx, off, 32));
  __shared__ float wmx[8], wsm[8], gmaxS;
  if ((tid & 31) == 0) wmx[tid >> 5] = mx;
  __syncthreads();
  if (tid == 0) {
    float g = wmx[0];
#pragma unroll
    for (int w = 1; w < 8; ++w) g = fmaxf(g, wmx[w]);
    gmaxS = g;
  }
  __syncthreads();
  float g = gmaxS;
  float s = 0.f;
#pragma unroll
  for (int i = 0; i < 8; ++i) s += __expf((mu[i] - g) * ieta);
#pragma unroll
  for (int off = 16; off > 0; off >>= 1) s += __shfl_xor(s, off, 32);
  if ((tid & 31) == 0) wsm[tid >> 5] = s;
  __syncthreads();
  if (tid == 0) {
    float Z = 0.f;
#pragma unroll
    for (int w = 0; w < 8; ++w) Z += wsm[w];
    maxu_ws[bm] = g;
    Z_ws[bm] = Z;
    float inc = g + logf(Z);
    incl_out[bm] = fminf(fmaxf(inc, -100.f), 100.f);
  }
}

// ---- K3.5: nest softmax over M ---------------------------------------------
__global__ void k_nest_probs(const float* __restrict__ ns,
                             const float* __restrict__ incl,
                             const float* __restrict__ eta,
                             float* __restrict__ np) {
  int b = blockIdx.x * blockDim.x + threadIdx.x;
  if (b >= BB) return;
  float lg[MM];
  float mx = -INFINITY;
#pragma unroll
  for (int m = 0; m < MM; ++m) {
    float v = ns[b * MM + m] + eta[m] * incl[b * MM + m];
    v = fminf(fmaxf(v, -50.f), 50.f);
    lg[m] = v;
    mx = fmaxf(mx, v);
  }
  float s = 0.f;
#pragma unroll
  for (int m = 0; m < MM; ++m) {
    lg[m] = __expf(lg[m] - mx);
    s += lg[m];
  }
  float inv = 1.f / s;
#pragma unroll
  for (int m = 0; m < MM; ++m) np[b * MM + m] = lg[m] * inv;
}

// ---- K4: fused task-probs softmax + joint einsum stream -------------------
__global__ __launch_bounds__(256) void k_probs(
    const float* __restrict__ util, const int* __restrict__ mask,
    const float* __restrict__ np, const float* __restrict__ eta,
    const float* __restrict__ maxu, const float* __restrict__ Z,
    float* __restrict__ tp, float* __restrict__ joint) {
  int b = blockIdx.x;
  __shared__ float snp[MM], smx[MM], sieta[MM], srz[MM];
  int tid = threadIdx.x;
  if (tid < MM) {
    snp[tid] = np[b * MM + tid];
    smx[tid] = maxu[b * MM + tid];
    sieta[tid] = 1.f / eta[tid];
    srz[tid] = 1.f / Z[b * MM + tid];
  }
  __syncthreads();
  const float* urow = util + (size_t)b * TT;
#pragma unroll 1
  for (int t = tid; t < TT; t += 256) {
    if (t + 256 < TT)
      __builtin_prefetch(&mask[(size_t)(b * MM) * TT + t + 256], 0, 1);
    float u = urow[t];
    float j = 0.f;
#pragma unroll
    for (int m = 0; m < MM; ++m) {
      int msk = mask[(size_t)(b * MM + m) * TT + t];
      float muv = msk ? u : NEGV;
      float p = __expf((muv - smx[m]) * sieta[m]) * srz[m];
      tp[(size_t)(b * MM + m) * TT + t] = p;
      j += msk ? snp[m] * p : 0.f;
    }
    joint[(size_t)b * TT + t] = j;
  }
}

// ---- host launcher ---------------------------------------------------------
extern "C" void kernel_launch(void* const* d_in, const int* in_sizes, int n_in,
                              void* d_out, int out_size, void* d_ws,
                              size_t ws_size, hipStream_t stream) {
  const float* state = (const float*)d_in[0];
  const float* tf    = (const float*)d_in[1];
  const int*   nmask = (const int*)d_in[2];
  const float* Wte = (const float*)d_in[3],  *bte = (const float*)d_in[4];
  const float* Wu1 = (const float*)d_in[5],  *bu1 = (const float*)d_in[6];
  const float* Wu2 = (const float*)d_in[7],  *bu2 = (const float*)d_in[8];
  const float* Wu3 = (const float*)d_in[9],  *bu3 = (const float*)d_in[10];
  const float* Wg1 = (const float*)d_in[11], *bg1 = (const float*)d_in[12];
  const float* Wg2 = (const float*)d_in[13], *bg2 = (const float*)d_in[14];
  const float* Wa  = (const float*)d_in[15], *ba  = (const float*)d_in[16];
  const float* Wn1 = (const float*)d_in[17], *bn1 = (const float*)d_in[18];
  const float* Wn2 = (const float*)d_in[19], *bn2 = (const float*)d_in[20];
  const float* Wn3 = (const float*)d_in[21], *bn3 = (const float*)d_in[22];
  const float* lam = (const float*)d_in[23];

  // outputs, concatenated flat in reference return order
  float* o_util  = (float*)d_out;                 // [B,T]   1048576
  float* o_np    = o_util + BB * TT;              // [B,M]   8192
  float* o_tp    = o_np + BB * MM;                // [B,M,T] 16777216
  float* o_joint = o_tp + (size_t)BB * MM * TT;   // [B,T]   1048576
  float* o_eta   = o_joint + BB * TT;             // [M]     16
  float* o_incl  = o_eta + MM;                    // [B,M]   8192
  float* o_ns    = o_incl + BB * MM;              // [B,M]   8192

  // workspace
  float* w_gctx = (float*)d_ws;                   // [B,E]     16384
  float* w_pool = w_gctx + BB * EE;               // [B,M,E]   262144
  float* w_maxu = w_pool + BB * MM * EE;          // [B,M]     8192
  float* w_Z    = w_maxu + BB * MM;               // [B,M]     8192

  k_eta<<<1, 32, 0, stream>>>(lam, o_eta);
  k_utilities<<<4096, 64, 0, stream>>>(state, tf, Wte, bte, Wu1, bu1, Wu2, bu2,
                                       Wu3, bu3, o_util);
  k_gctx<<<16, 64, 0, stream>>>(state, Wg1, bg1, Wg2, bg2, w_gctx);
  k_pool<<<BB * MM, 256, 0, stream>>>(tf, Wte, bte, nmask, Wa, ba, w_pool);
  k_nest_scores<<<BB / 2, 64, 0, stream>>>(w_gctx, w_pool, Wn1, bn1, Wn2, bn2,
                                           Wn3, bn3, o_ns);
  k_inclusive<<<BB * MM, 256, 0, stream>>>(o_util, nmask, o_eta, w_maxu, w_Z,
                                           o_incl);
  k_nest_probs<<<(BB + 127) / 128, 128, 0, stream>>>(o_ns, o_incl, o_eta, o_np);
  k_probs<<<BB, 256, 0, stream>>>(o_util, nmask, o_np, o_eta, w_maxu, w_Z,
                                  o_tp, o_joint);
}